// GCN_77704548319407
// MI455X (gfx1250) — compile-verified
//
#include <hip/hip_runtime.h>

// ---------------------------------------------------------------------------
// GCN (2x GCNConv + mean pool + linear head) for MI455X / gfx1250 (wave32).
// Dense projections use V_WMMA_F32_16X16X4_F32 (fp32-exact matrix cores).
// Edge aggregation uses native fp32 global atomics (L2-resident, 192MB L2).
// ---------------------------------------------------------------------------

#define HID 64   // hidden width (both layers output 64 features)

typedef __attribute__((ext_vector_type(2))) float v2f;
typedef __attribute__((ext_vector_type(8))) float v8f;

__device__ __forceinline__ void atomAddF(float* p, float v) {
    // lowers to global_atomic_add_f32 (no CAS loop)
    unsafeAtomicAdd(p, v);
}

// ---------------- degree / normalization ----------------

__global__ void k_deg_init(float* __restrict__ deg, int n) {
    int i = blockIdx.x * blockDim.x + threadIdx.x;
    if (i < n) deg[i] = 1.0f;            // self-loop contributes 1
}

__global__ void k_deg_count(const int* __restrict__ dst, float* __restrict__ deg, int e) {
    int i = blockIdx.x * blockDim.x + threadIdx.x;
    if (i < e) atomAddF(&deg[dst[i]], 1.0f);
}

__global__ void k_deg_to_dinv(float* __restrict__ deg, int n) {
    int i = blockIdx.x * blockDim.x + threadIdx.x;
    if (i < n) {
        float d = deg[i];
        deg[i] = (d > 0.0f) ? rsqrtf(d) : 0.0f;   // in-place: deg -> dinv
    }
}

// ---------------- WMMA fp32 GEMM:  Hout[n x 64] = X[n x K] @ W[K x 64] -----
// grid.x = n/16 row tiles, block = 128 threads (4 waves), wave w owns cols
// [16w, 16w+16). K-loop uses V_WMMA_F32_16X16X4_F32 (A 16x4, B 4x16, C 16x16).
// A layout  : lanes 0-15 hold row M=lane, (K0,K1); lanes 16-31 same row, (K2,K3).
// B layout  : lanes 0-15 hold col N=lane, rows (K0,K1); lanes 16-31 rows (K2,K3).
// C/D layout: VGPR r -> M = r (lanes 0-15) / M = r+8 (lanes 16-31), N = lane%16.

template <int K>
__global__ __launch_bounds__(128)
void k_gemm_wmma_f32(const float* __restrict__ X,
                     const float* __restrict__ W,
                     float* __restrict__ Hout) {
    const int lane = threadIdx.x & 31;
    const int wave = threadIdx.x >> 5;
    const int half = lane >> 4;      // 0: lanes 0-15, 1: lanes 16-31
    const int lm   = lane & 15;
    const int row0 = blockIdx.x * 16;
    const int col0 = wave * 16;

    v8f c = {0.f, 0.f, 0.f, 0.f, 0.f, 0.f, 0.f, 0.f};

    const float* xrow = X + (size_t)(row0 + lm) * K;
    const float* wcol = W + col0 + lm;

#pragma unroll 4
    for (int k0 = 0; k0 < K; k0 += 4) {
        const int ka = k0 + 2 * half;
        v2f a, b;
        a.x = xrow[ka];
        a.y = xrow[ka + 1];
        b.x = wcol[(size_t)ka * HID];
        b.y = wcol[(size_t)(ka + 1) * HID];
        // 8 args: (neg_a, A, neg_b, B, c_mod, C, reuse_a, reuse_b)
        c = __builtin_amdgcn_wmma_f32_16x16x4_f32(
                false, a, false, b, (short)0, c, false, false);
    }

#pragma unroll
    for (int r = 0; r < 8; ++r) {
        const int m = row0 + r + 8 * half;
        Hout[(size_t)m * HID + col0 + lm] = c[r];
    }
}

// ---------------- aggregation ----------------

// agg[i][:] = dinv[i]^2 * h[i][:]   (self-loop term, initializes accumulator)
__global__ void k_agg_init(const float* __restrict__ h, const float* __restrict__ dinv,
                           float* __restrict__ agg, int n) {
    int idx = blockIdx.x * blockDim.x + threadIdx.x;      // over n * 16
    int i  = idx >> 4;
    if (i >= n) return;
    int fb = (idx & 15) << 2;
    float di = dinv[i];
    float s  = di * di;
    const float4 hv = *(const float4*)(h + (size_t)i * HID + fb);
    float4 o;
    o.x = s * hv.x; o.y = s * hv.y; o.z = s * hv.z; o.w = s * hv.w;
    *(float4*)(agg + (size_t)i * HID + fb) = o;
}

// agg[dst][:] += dinv[src]*dinv[dst] * h[src][:]   (16 threads per edge)
__global__ void k_edge_scatter(const int* __restrict__ src, const int* __restrict__ dst,
                               const float* __restrict__ dinv, const float* __restrict__ h,
                               float* __restrict__ agg, int e) {
    long long idx = (long long)blockIdx.x * blockDim.x + threadIdx.x;  // over e * 16
    int ei = (int)(idx >> 4);
    if (ei >= e) return;
    int fb = ((int)idx & 15) << 2;
    int s = src[ei];
    int d = dst[ei];
    float nrm = dinv[s] * dinv[d];
    const float4 hv = *(const float4*)(h + (size_t)s * HID + fb);
    float* o = agg + (size_t)d * HID + fb;
    atomAddF(o + 0, nrm * hv.x);
    atomAddF(o + 1, nrm * hv.y);
    atomAddF(o + 2, nrm * hv.z);
    atomAddF(o + 3, nrm * hv.w);
}

// z = relu(agg + b), in place
__global__ void k_bias_relu(float* __restrict__ agg, const float* __restrict__ b, int n) {
    int idx = blockIdx.x * blockDim.x + threadIdx.x;      // over n * 16
    int i  = idx >> 4;
    if (i >= n) return;
    int fb = (idx & 15) << 2;
    float4 v  = *(const float4*)(agg + (size_t)i * HID + fb);
    float4 bv = *(const float4*)(b + fb);
    v.x = fmaxf(v.x + bv.x, 0.f);
    v.y = fmaxf(v.y + bv.y, 0.f);
    v.z = fmaxf(v.z + bv.z, 0.f);
    v.w = fmaxf(v.w + bv.w, 0.f);
    *(float4*)(agg + (size_t)i * HID + fb) = v;
}

// ---------------- pooling + head ----------------

__global__ void k_zero(float* __restrict__ p, int n) {
    int i = blockIdx.x * blockDim.x + threadIdx.x;
    if (i < n) p[i] = 0.f;
}

__global__ void k_pool_scatter(const float* __restrict__ z, const int* __restrict__ batch,
                               float* __restrict__ pooled, float* __restrict__ cnt, int n) {
    int idx = blockIdx.x * blockDim.x + threadIdx.x;      // over n * 16
    int i  = idx >> 4;
    if (i >= n) return;
    int fb = (idx & 15) << 2;
    int g  = batch[i];
    const float4 v = *(const float4*)(z + (size_t)i * HID + fb);
    float* o = pooled + (size_t)g * HID + fb;
    atomAddF(o + 0, v.x);
    atomAddF(o + 1, v.y);
    atomAddF(o + 2, v.z);
    atomAddF(o + 3, v.w);
    if (fb == 0) atomAddF(&cnt[g], 1.0f);
}

// one wave (32 lanes) per graph: out[g] = (pooled[g]/max(cnt,1)) . Wh + bh
__global__ void k_head(const float* __restrict__ pooled, const float* __restrict__ cnt,
                       const float* __restrict__ Wh, const float* __restrict__ bh,
                       float* __restrict__ out) {
    int g    = blockIdx.x;
    int lane = threadIdx.x;   // 0..31
    const float* pg = pooled + (size_t)g * HID;
    float s = pg[lane] * Wh[lane] + pg[lane + 32] * Wh[lane + 32];
#pragma unroll
    for (int off = 16; off > 0; off >>= 1)
        s += __shfl_xor(s, off, 32);
    if (lane == 0)
        out[g] = s / fmaxf(cnt[g], 1.0f) + bh[0];
}

// ---------------- launch ----------------

extern "C" void kernel_launch(void* const* d_in, const int* in_sizes, int n_in,
                              void* d_out, int out_size, void* d_ws, size_t ws_size,
                              hipStream_t stream) {
    const float* x     = (const float*)d_in[0];   // [N, 128]
    const int*   eidx  = (const int*)  d_in[1];   // [2, E]
    const int*   batch = (const int*)  d_in[2];   // [N]
    const float* W1    = (const float*)d_in[3];   // [128, 64]
    const float* b1    = (const float*)d_in[4];   // [64]
    const float* W2    = (const float*)d_in[5];   // [64, 64]
    const float* b2    = (const float*)d_in[6];   // [64]
    const float* Wh    = (const float*)d_in[7];   // [64, 1]
    const float* bh    = (const float*)d_in[8];   // [1]
    float*       out   = (float*)d_out;           // [G]

    const int n = in_sizes[2];        // N nodes
    const int e = in_sizes[1] / 2;    // E edges
    const int g = out_size;           // G graphs

    const int* src = eidx;
    const int* dst = eidx + e;

    // workspace layout (floats): dinv[n] | hbuf[n*64] | abuf[n*64] | pooled[g*64] | cnt[g]
    float* ws     = (float*)d_ws;
    float* dinv   = ws;
    float* hbuf   = dinv + n;
    float* abuf   = hbuf + (size_t)n * HID;
    float* pooled = abuf + (size_t)n * HID;
    float* cnt    = pooled + (size_t)g * HID;
    (void)ws_size; (void)n_in; (void)cnt;

    const int T   = 256;
    const int gN  = (n + T - 1) / T;
    const int gE  = (e + T - 1) / T;
    const int gNF = (n * 16 + T - 1) / T;                 // n*64 feats / 4-wide
    const int gEF = (int)(((long long)e * 16 + T - 1) / T);

    // symmetric-normalization coefficients
    k_deg_init   <<<gN, T, 0, stream>>>(dinv, n);
    k_deg_count  <<<gE, T, 0, stream>>>(dst, dinv, e);
    k_deg_to_dinv<<<gN, T, 0, stream>>>(dinv, n);

    // ---- layer 1: h1 = X @ W1 ; agg = D^-1/2 A D^-1/2 h1 ; z1 = relu(agg+b1)
    k_gemm_wmma_f32<128><<<n / 16, 128, 0, stream>>>(x, W1, hbuf);
    k_agg_init    <<<gNF, T, 0, stream>>>(hbuf, dinv, abuf, n);
    k_edge_scatter<<<gEF, T, 0, stream>>>(src, dst, dinv, hbuf, abuf, e);
    k_bias_relu   <<<gNF, T, 0, stream>>>(abuf, b1, n);   // z1 in abuf

    // ---- layer 2: h2 = z1 @ W2 ; same aggregation ; z2 = relu(agg+b2)
    k_gemm_wmma_f32<64><<<n / 16, 128, 0, stream>>>(abuf, W2, hbuf);
    k_agg_init    <<<gNF, T, 0, stream>>>(hbuf, dinv, abuf, n);  // overwrites z1 (done with it)
    k_edge_scatter<<<gEF, T, 0, stream>>>(src, dst, dinv, hbuf, abuf, e);
    k_bias_relu   <<<gNF, T, 0, stream>>>(abuf, b2, n);   // z2 in abuf

    // ---- global mean pool + linear head
    k_zero        <<<(g * HID + g + T - 1) / T, T, 0, stream>>>(pooled, g * HID + g);
    k_pool_scatter<<<gNF, T, 0, stream>>>(abuf, batch, pooled, cnt, n);
    k_head        <<<g, 32, 0, stream>>>(pooled, cnt, Wh, bh, out);
}